// ProtoNet_42494406427173
// MI455X (gfx1250) — compile-verified
//
#include <hip/hip_runtime.h>
#include <hip/hip_bf16.h>
#include <math.h>

typedef __attribute__((ext_vector_type(2))) float v2f;
typedef __attribute__((ext_vector_type(8))) float v8f;

#define GAMMA_EMA 0.999f
#define KPROT     5
#define CCLS      201
#define SINK_EPS  0.05f
#define LNEPS     1e-5f

#define NPATCH  32768
#define DDIM    256
#define BDDIM   768
#define NPL     1005      // C*K

// ---- d_out float offsets ----
#define OUT_SEG 0
#define OUT_PL  6586368
#define OUT_PPL 39518208
#define OUT_TGT 72450048
#define OUT_NP  72482816

// ---- workspace float offsets ----
#define WS_PATCH 0
#define WS_PROTN 8388608
#define WS_W     8645888
#define WS_CN    8809728
#define WS_ACK   8842496
#define WS_TCK   8843520
#define WS_S     8844544
#define WS_BN    8844800
#define WS_NCK   8845056
#define WS_FCK   8846080
#define WS_CORR  9103360

#define NEG_BIG -3.402823466e+38f

__device__ __forceinline__ float wave_sum(float v) {
    #pragma unroll
    for (int m = 16; m >= 1; m >>= 1) v += __shfl_xor(v, m, 32);
    return v;
}

#define WMMA_F32(A_, B_, C_) \
    __builtin_amdgcn_wmma_f32_16x16x4_f32(false, (A_), false, (B_), (short)0, (C_), false, false)

// ================= GEMM1: f = tokens[32768x768] @ proj_w[768x256] + b =================
// WG = 8 waves (4x2), 128x64 block tile; 32x32 register tile per wave (4 accumulators).
__global__ __launch_bounds__(256)
void gemm1_kernel(const float* __restrict__ A, const float* __restrict__ B,
                  const float* __restrict__ bias, float* __restrict__ Cout) {
    __shared__ float sA[128][36];   // stride 36: conflict-free frag reads, 16B-aligned rows
    __shared__ float sB[32][72];    // stride 72: K/K+2 half-waves hit disjoint bank sets
    const int tid = threadIdx.x;
    const int wid = tid >> 5, lane = tid & 31;
    const int wm = wid & 3, wn = wid >> 2;
    const int m0 = blockIdx.x * 128;
    const int n0 = blockIdx.y * 64;

    v8f acc00 = {}, acc01 = {}, acc10 = {}, acc11 = {};
    for (int k0 = 0; k0 < BDDIM; k0 += 32) {
        // stage A 128x32 as float4 (1024 quads / 256 threads)
        #pragma unroll
        for (int i = 0; i < 4; ++i) {
            int idx = tid + i * 256, r = idx >> 3, q = idx & 7;
            float4 v = *(const float4*)&A[(size_t)(m0 + r) * BDDIM + k0 + q * 4];
            *(float4*)&sA[r][q * 4] = v;
        }
        // stage B 32x64 as float4 (512 quads / 256 threads)
        #pragma unroll
        for (int i = 0; i < 2; ++i) {
            int idx = tid + i * 256, r = idx >> 4, q = idx & 15;
            float4 v = *(const float4*)&B[(size_t)(k0 + r) * DDIM + n0 + q * 4];
            *(float4*)&sB[r][q * 4] = v;
        }
        __syncthreads();
        if (k0 + 32 < BDDIM) {   // hint-prefetch next A chunk -> global_prefetch_b8
            int r = tid >> 3, q = tid & 7;
            __builtin_prefetch(&A[(size_t)(m0 + r) * BDDIM + k0 + 32 + q * 4], 0, 1);
        }
        const int mr = wm * 32 + (lane & 15);
        const int nc = wn * 32 + (lane & 15);
        const int ks = (lane >> 4) << 1;      // lanes 0-15 -> K0/K1, 16-31 -> K2/K3
        #pragma unroll
        for (int kk = 0; kk < 32; kk += 4) {
            v2f a0, a1, b0, b1;
            a0.x = sA[mr][kk + ks];        a0.y = sA[mr][kk + ks + 1];
            a1.x = sA[mr + 16][kk + ks];   a1.y = sA[mr + 16][kk + ks + 1];
            b0.x = sB[kk + ks][nc];        b0.y = sB[kk + ks + 1][nc];
            b1.x = sB[kk + ks][nc + 16];   b1.y = sB[kk + ks + 1][nc + 16];
            acc00 = WMMA_F32(a0, b0, acc00);
            acc01 = WMMA_F32(a0, b1, acc01);
            acc10 = WMMA_F32(a1, b0, acc10);
            acc11 = WMMA_F32(a1, b1, acc11);
        }
        __syncthreads();
    }
    const int colb = n0 + wn * 32 + (lane & 15);
    const int rowb = m0 + wm * 32 + ((lane >> 4) << 3);
    const float bv0 = bias[colb], bv1 = bias[colb + 16];
    #pragma unroll
    for (int r = 0; r < 8; ++r) {
        Cout[(size_t)(rowb + r) * DDIM + colb]           = acc00[r] + bv0;
        Cout[(size_t)(rowb + r) * DDIM + colb + 16]      = acc01[r] + bv1;
        Cout[(size_t)(rowb + 16 + r) * DDIM + colb]      = acc10[r] + bv0;
        Cout[(size_t)(rowb + 16 + r) * DDIM + colb + 16] = acc11[r] + bv1;
    }
}

// ======= GEMM2: X = patches[32768x256] @ protosn^T[256x1005]; dual-layout epilogue =======
__global__ __launch_bounds__(256)
void gemm2_kernel(const float* __restrict__ A, const float* __restrict__ Bt,
                  float* __restrict__ out_pl, float* __restrict__ out_ppl) {
    __shared__ float sA[128][36];
    __shared__ float sB[32][72];
    const int tid = threadIdx.x;
    const int wid = tid >> 5, lane = tid & 31;
    const int wm = wid & 3, wn = wid >> 2;
    const int m0 = blockIdx.x * 128;
    const int n0 = blockIdx.y * 64;

    v8f acc00 = {}, acc01 = {}, acc10 = {}, acc11 = {};
    for (int k0 = 0; k0 < DDIM; k0 += 32) {
        #pragma unroll
        for (int i = 0; i < 4; ++i) {
            int idx = tid + i * 256, r = idx >> 3, q = idx & 7;
            float4 v = *(const float4*)&A[(size_t)(m0 + r) * DDIM + k0 + q * 4];
            *(float4*)&sA[r][q * 4] = v;
        }
        // B is accessed transposed: contiguous along k; scatter 4 scalars into LDS rows
        #pragma unroll
        for (int i = 0; i < 2; ++i) {
            int idx = tid + i * 256, q = idx & 7, j = idx >> 3;   // q: k-quad, j: 0..63
            int jj = n0 + j;
            float4 v = {0.f, 0.f, 0.f, 0.f};
            if (jj < NPL) v = *(const float4*)&Bt[(size_t)jj * DDIM + k0 + q * 4];
            sB[q * 4 + 0][j] = v.x;
            sB[q * 4 + 1][j] = v.y;
            sB[q * 4 + 2][j] = v.z;
            sB[q * 4 + 3][j] = v.w;
        }
        __syncthreads();
        if (k0 + 32 < DDIM) {
            int r = tid >> 3, q = tid & 7;
            __builtin_prefetch(&A[(size_t)(m0 + r) * DDIM + k0 + 32 + q * 4], 0, 1);
        }
        const int mr = wm * 32 + (lane & 15);
        const int nc = wn * 32 + (lane & 15);
        const int ks = (lane >> 4) << 1;
        #pragma unroll
        for (int kk = 0; kk < 32; kk += 4) {
            v2f a0, a1, b0, b1;
            a0.x = sA[mr][kk + ks];        a0.y = sA[mr][kk + ks + 1];
            a1.x = sA[mr + 16][kk + ks];   a1.y = sA[mr + 16][kk + ks + 1];
            b0.x = sB[kk + ks][nc];        b0.y = sB[kk + ks + 1][nc];
            b1.x = sB[kk + ks][nc + 16];   b1.y = sB[kk + ks + 1][nc + 16];
            acc00 = WMMA_F32(a0, b0, acc00);
            acc01 = WMMA_F32(a0, b1, acc01);
            acc10 = WMMA_F32(a1, b0, acc10);
            acc11 = WMMA_F32(a1, b1, acc11);
        }
        __syncthreads();
    }
    const int j0 = n0 + wn * 32 + (lane & 15);
    const int j1 = j0 + 16;
    const int rowb = m0 + wm * 32 + ((lane >> 4) << 3);
    const int cc0 = j0 / KPROT, kk0 = j0 % KPROT;
    const int cc1 = j1 / KPROT, kk1 = j1 % KPROT;
    #pragma unroll
    for (int r = 0; r < 8; ++r) {
        size_t row0 = (size_t)(rowb + r), row1 = (size_t)(rowb + 16 + r);
        if (j0 < NPL) {
            out_pl [row0 * NPL + j0] = acc00[r];
            out_ppl[row0 * NPL + (size_t)kk0 * CCLS + cc0] = acc00[r];
            out_pl [row1 * NPL + j0] = acc10[r];
            out_ppl[row1 * NPL + (size_t)kk0 * CCLS + cc0] = acc10[r];
        }
        if (j1 < NPL) {
            out_pl [row0 * NPL + j1] = acc01[r];
            out_ppl[row0 * NPL + (size_t)kk1 * CCLS + cc1] = acc01[r];
            out_pl [row1 * NPL + j1] = acc11[r];
            out_ppl[row1 * NPL + (size_t)kk1 * CCLS + cc1] = acc11[r];
        }
    }
}

// ================= LayerNorm + L2 normalize, in place, one wave per 256-wide row =========
__global__ __launch_bounds__(256)
void ln_l2_kernel(float* __restrict__ X, const float* __restrict__ g,
                  const float* __restrict__ b, int nrows) {
    const int wid = threadIdx.x >> 5, lane = threadIdx.x & 31;
    const int row = blockIdx.x * 8 + wid;
    if (row >= nrows) return;
    float* x = X + (size_t)row * DDIM;
    float v[8], s = 0.f, ss = 0.f;
    #pragma unroll
    for (int i = 0; i < 8; ++i) { v[i] = x[lane + i * 32]; s += v[i]; ss += v[i] * v[i]; }
    s = wave_sum(s); ss = wave_sum(ss);
    const float mu = s * (1.0f / DDIM);
    const float var = ss * (1.0f / DDIM) - mu * mu;
    const float rstd = rsqrtf(var + LNEPS);
    float n2 = 0.f;
    #pragma unroll
    for (int i = 0; i < 8; ++i) {
        float y = (v[i] - mu) * rstd * g[lane + i * 32] + b[lane + i * 32];
        v[i] = y; n2 += y * y;
    }
    n2 = wave_sum(n2);
    const float inv = 1.0f / fmaxf(sqrtf(n2), 1e-12f);
    #pragma unroll
    for (int i = 0; i < 8; ++i) x[lane + i * 32] = v[i] * inv;
}

// ================= plain L2 normalize rows (prototypes) =================
__global__ __launch_bounds__(256)
void l2_kernel(const float* __restrict__ S, float* __restrict__ Dst, int nrows) {
    const int wid = threadIdx.x >> 5, lane = threadIdx.x & 31;
    const int row = blockIdx.x * 8 + wid;
    if (row >= nrows) return;
    const float* x = S + (size_t)row * DDIM;
    float v[8], n2 = 0.f;
    #pragma unroll
    for (int i = 0; i < 8; ++i) { v[i] = x[lane + i * 32]; n2 += v[i] * v[i]; }
    n2 = wave_sum(n2);
    const float inv = 1.0f / fmaxf(sqrtf(n2), 1e-12f);
    #pragma unroll
    for (int i = 0; i < 8; ++i) Dst[(size_t)row * DDIM + lane + i * 32] = v[i] * inv;
}

// ========== pcl = LN(max_k ppl) -> seg, preds/correct, w = exp(q_gt/eps), S_c, Bn ==========
__global__ __launch_bounds__(256)
void pcl_kernel(const float* __restrict__ pl, const int* __restrict__ gt,
                const float* __restrict__ cng, const float* __restrict__ cnb,
                float* __restrict__ seg, float* __restrict__ corr,
                float* __restrict__ w, float* __restrict__ Ssum, float* __restrict__ Bn) {
    const int wid = threadIdx.x >> 5, lane = threadIdx.x & 31;
    const int n = blockIdx.x * 8 + wid;
    if (n >= NPATCH) return;
    const float* row = pl + (size_t)n * NPL;
    float mx[7], s = 0.f, ss = 0.f;
    #pragma unroll
    for (int i = 0; i < 7; ++i) {
        int c = lane + i * 32;
        float m = NEG_BIG;
        if (c < CCLS) {
            const float* p = row + c * KPROT;
            m = p[0];
            #pragma unroll
            for (int k = 1; k < KPROT; ++k) m = fmaxf(m, p[k]);
            s += m; ss += m * m;
        }
        mx[i] = m;
    }
    s = wave_sum(s); ss = wave_sum(ss);
    const float mu = s * (1.0f / CCLS);
    const float var = ss * (1.0f / CCLS) - mu * mu;
    const float rstd = rsqrtf(var + LNEPS);
    const int bb = n >> 10, hw = n & 1023;
    float best = NEG_BIG; int bidx = CCLS;
    #pragma unroll
    for (int i = 0; i < 7; ++i) {
        int c = lane + i * 32;
        if (c < CCLS) {
            float y = (mx[i] - mu) * rstd * cng[c] + cnb[c];
            seg[((size_t)bb * CCLS + c) * 1024 + hw] = y;
            if (y > best) { best = y; bidx = c; }   // within-lane later c is larger index
        }
    }
    #pragma unroll
    for (int m = 16; m >= 1; m >>= 1) {
        float ov = __shfl_xor(best, m, 32);
        int   oi = __shfl_xor(bidx, m, 32);
        if (ov > best || (ov == best && oi < bidx)) { best = ov; bidx = oi; }
    }
    const int g = gt[n];
    if (lane == 0) {
        corr[n] = (g == bidx) ? 1.0f : 0.0f;
        float sw = 0.f;
        #pragma unroll
        for (int k = 0; k < KPROT; ++k) {
            float e = expf(row[g * KPROT + k] * (1.0f / SINK_EPS));
            w[(size_t)n * KPROT + k] = e; sw += e;
        }
        atomicAdd(&Ssum[g], sw);
        atomicAdd(&Bn[g], 1.0f);
    }
}

// ================= init / small helpers =================
__global__ void init_kernel(float* Ssum, float* Bn, float* nck, float* fck, float* Cn) {
    int i = blockIdx.x * 256 + threadIdx.x;
    if (i < CCLS) { Ssum[i] = 0.f; Bn[i] = 0.f; }
    if (i < NPL) nck[i] = 0.f;
    if (i < NPL * DDIM) fck[i] = 0.f;
    if (i < NPATCH) Cn[i] = 1.0f;
}
__global__ void initA_kernel(const float* Ssum, float* A) {
    int i = blockIdx.x * 256 + threadIdx.x;
    if (i < NPL) A[i] = 1.0f / fmaxf(Ssum[i / KPROT], 1e-30f);
}
__global__ void zeroT_kernel(float* t) {
    int i = blockIdx.x * 256 + threadIdx.x;
    if (i < NPL) t[i] = 0.f;
}
__global__ void accumT_kernel(const int* __restrict__ gt, const float* __restrict__ w,
                              const float* __restrict__ Cn, float* __restrict__ t) {
    int n = blockIdx.x * 256 + threadIdx.x;
    if (n >= NPATCH) return;
    int g = gt[n]; float cn = Cn[n];
    #pragma unroll
    for (int k = 0; k < KPROT; ++k)
        atomicAdd(&t[g * KPROT + k], w[(size_t)n * KPROT + k] * cn);
}
__global__ void updateA_kernel(float* A, const float* t) {
    int i = blockIdx.x * 256 + threadIdx.x;
    if (i >= NPL) return;
    float a = A[i];
    float rs = fmaxf(a * t[i], 1e-30f);
    A[i] = a / (rs * (float)KPROT);
}
__global__ void updateC_kernel(const int* __restrict__ gt, const float* __restrict__ w,
                               const float* __restrict__ A, const float* __restrict__ Bn,
                               float* __restrict__ Cn) {
    int n = blockIdx.x * 256 + threadIdx.x;
    if (n >= NPATCH) return;
    int g = gt[n];
    float u = 0.f;
    #pragma unroll
    for (int k = 0; k < KPROT; ++k) u += w[(size_t)n * KPROT + k] * A[g * KPROT + k];
    float c = Cn[n];
    float col = fmaxf(c * u, 1e-30f);
    Cn[n] = c / (col * fmaxf(Bn[g], 1.0f));
}

// ===== hard assignment: idx=argmax_k w*A, proto_target, scatter f_ck / n_ck =====
__global__ __launch_bounds__(256)
void assign_kernel(const int* __restrict__ gt, const float* __restrict__ w,
                   const float* __restrict__ A, const float* __restrict__ corr,
                   const float* __restrict__ patches, float* __restrict__ tgt,
                   float* __restrict__ nck, float* __restrict__ fck) {
    const int wid = threadIdx.x >> 5, lane = threadIdx.x & 31;
    const int n = blockIdx.x * 8 + wid;
    if (n >= NPATCH) return;
    const int g = gt[n];
    float bv = NEG_BIG; int bi = 0;
    #pragma unroll
    for (int k = 0; k < KPROT; ++k) {
        float v = w[(size_t)n * KPROT + k] * A[g * KPROT + k];
        if (v > bv) { bv = v; bi = k; }
    }
    if (lane == 0) tgt[n] = (float)(bi + KPROT * g);
    if (corr[n] != 0.0f) {
        if (lane == 0) atomicAdd(&nck[g * KPROT + bi], 1.0f);
        const float* p = patches + (size_t)n * DDIM;
        float* f = fck + (size_t)(g * KPROT + bi) * DDIM;
        #pragma unroll
        for (int i = 0; i < 8; ++i) atomicAdd(&f[lane + i * 32], p[lane + i * 32]);
    }
}

// ===== EMA update + renormalize prototypes =====
__global__ __launch_bounds__(256)
void newproto_kernel(const float* __restrict__ protosn, const float* __restrict__ fck,
                     const float* __restrict__ nck, float* __restrict__ out) {
    const int wid = threadIdx.x >> 5, lane = threadIdx.x & 31;
    const int ck = blockIdx.x * 8 + wid;
    if (ck >= NPL) return;
    const int c = ck / KPROT;
    float csum = 0.f;
    #pragma unroll
    for (int k = 0; k < KPROT; ++k) csum += nck[c * KPROT + k];
    const bool cond = (csum > 0.0f) && (nck[ck] != 0.0f);
    const float* pr = protosn + (size_t)ck * DDIM;
    const float* f  = fck + (size_t)ck * DDIM;
    float v[8], f2 = 0.f;
    #pragma unroll
    for (int i = 0; i < 8; ++i) { v[i] = f[lane + i * 32]; f2 += v[i] * v[i]; }
    f2 = wave_sum(f2);
    const float finv = 1.0f / fmaxf(sqrtf(f2), 1e-12f);
    float o[8], n2 = 0.f;
    #pragma unroll
    for (int i = 0; i < 8; ++i) {
        float p = pr[lane + i * 32];
        float x = cond ? (GAMMA_EMA * p + (1.0f - GAMMA_EMA) * v[i] * finv) : p;
        o[i] = x; n2 += x * x;
    }
    n2 = wave_sum(n2);
    const float inv = 1.0f / fmaxf(sqrtf(n2), 1e-12f);
    #pragma unroll
    for (int i = 0; i < 8; ++i) out[(size_t)ck * DDIM + lane + i * 32] = o[i] * inv;
}

extern "C" void kernel_launch(void* const* d_in, const int* in_sizes, int n_in,
                              void* d_out, int out_size, void* d_ws, size_t ws_size,
                              hipStream_t stream) {
    const float* tokens  = (const float*)d_in[0];
    const int*   gt      = (const int*)  d_in[1];
    const float* proj_w  = (const float*)d_in[2];
    const float* proj_b  = (const float*)d_in[3];
    const float* fng     = (const float*)d_in[4];
    const float* fnb     = (const float*)d_in[5];
    const float* protos  = (const float*)d_in[6];
    const float* cng     = (const float*)d_in[7];
    const float* cnb     = (const float*)d_in[8];

    float* out = (float*)d_out;
    float* seg = out + OUT_SEG;
    float* opl = out + OUT_PL;
    float* opp = out + OUT_PPL;
    float* otg = out + OUT_TGT;
    float* onp = out + OUT_NP;

    float* ws      = (float*)d_ws;
    float* patches = ws + WS_PATCH;
    float* protn   = ws + WS_PROTN;
    float* w       = ws + WS_W;
    float* Cn      = ws + WS_CN;
    float* Ack     = ws + WS_ACK;
    float* tck     = ws + WS_TCK;
    float* Ssum    = ws + WS_S;
    float* Bn      = ws + WS_BN;
    float* nck     = ws + WS_NCK;
    float* fck     = ws + WS_FCK;
    float* corr    = ws + WS_CORR;

    init_kernel<<<1005, 256, 0, stream>>>(Ssum, Bn, nck, fck, Cn);

    gemm1_kernel<<<dim3(NPATCH / 128, DDIM / 64), 256, 0, stream>>>(tokens, proj_w, proj_b, patches);
    ln_l2_kernel<<<NPATCH / 8, 256, 0, stream>>>(patches, fng, fnb, NPATCH);
    l2_kernel<<<(NPL + 7) / 8, 256, 0, stream>>>(protos, protn, NPL);

    gemm2_kernel<<<dim3(NPATCH / 128, 16), 256, 0, stream>>>(patches, protn, opl, opp);

    pcl_kernel<<<NPATCH / 8, 256, 0, stream>>>(opl, gt, cng, cnb, seg, corr, w, Ssum, Bn);
    initA_kernel<<<4, 256, 0, stream>>>(Ssum, Ack);

    for (int it = 0; it < 3; ++it) {
        zeroT_kernel<<<4, 256, 0, stream>>>(tck);
        accumT_kernel<<<NPATCH / 256, 256, 0, stream>>>(gt, w, Cn, tck);
        updateA_kernel<<<4, 256, 0, stream>>>(Ack, tck);
        updateC_kernel<<<NPATCH / 256, 256, 0, stream>>>(gt, w, Ack, Bn, Cn);
    }

    assign_kernel<<<NPATCH / 8, 256, 0, stream>>>(gt, w, Ack, corr, patches, otg, nck, fck);
    newproto_kernel<<<(NPL + 7) / 8, 256, 0, stream>>>(protn, fck, nck, onp);
}